// FourierLayer_28733331210668
// MI455X (gfx1250) — compile-verified
//
#include <hip/hip_runtime.h>
#include <math.h>

// ---------------------------------------------------------------------------
// FourierLayer fused kernel for gfx1250 (MI455X).
//
// Math: pred = A · (m ⊛ (Aᵀ · X · A)) · Aᵀ  per (b,c) image, where
//   A[u,i]   = EX(p1d[u]-g1d[i])              (128x10 separable mollifier)
//   m[d]     = 1e-4 * Σ_k Dmult[k] cos(2π k·d/10)   (real spectral operator)
// Memory-bound problem: 256 MB total traffic -> ~11 us at 23.3 TB/s.
// All contractions use V_WMMA_F32_16X16X4_F32 on LDS-resident tiles; the
// image is staged HBM<->LDS with gfx1250 async LDS transfers (ASYNCcnt).
// ---------------------------------------------------------------------------

typedef float v2f __attribute__((ext_vector_type(2)));
typedef float v4f __attribute__((ext_vector_type(4)));
typedef float v8f __attribute__((ext_vector_type(8)));
typedef int   v4i __attribute__((ext_vector_type(4)));

typedef __attribute__((address_space(1))) v4i GV4i;  // global int4
typedef __attribute__((address_space(3))) v4i LV4i;  // LDS int4

#ifndef __has_builtin
#define __has_builtin(x) 0
#endif

#if __has_builtin(__builtin_amdgcn_global_load_async_to_lds_b128) && \
    __has_builtin(__builtin_amdgcn_global_store_async_from_lds_b128)
#define USE_ASYNC_LDS 1
#else
#define USE_ASYNC_LDS 0
#endif

#if __has_builtin(__builtin_amdgcn_s_wait_asynccnt)
#define WAIT_ASYNC0() __builtin_amdgcn_s_wait_asynccnt(0)
#else
#define WAIT_ASYNC0() asm volatile("s_wait_asynccnt 0x0" ::: "memory")
#endif

#define WMMA_F32(a, b, c) \
  __builtin_amdgcn_wmma_f32_16x16x4_f32(false, (a), false, (b), (short)0, (c), false, false)

static constexpr float C_L      = 50.265482457436692f;   // 16*pi
static constexpr float C_TAU    = 7.68f;                 // 12*(L/(2*pi*10))^2
static constexpr float C_PI     = 3.14159265358979323f;
static constexpr float C_I4TAU  = 1.0f / (4.0f * 7.68f);
static constexpr float C_K0     = 0.125f;                // 2*pi/L
static constexpr float C_W10    = 0.62831853071795865f;  // 2*pi/10

// --------------------------------------------------------------------------
// Setup kernel 1: A[128][16]  (cols 10..15 zero-padded), row-major in ws.
// --------------------------------------------------------------------------
__global__ void build_A_kernel(float* __restrict__ A)
{
  int tid = threadIdx.x;
  for (int idx = tid; idx < 128 * 16; idx += 256) {
    int u = idx >> 4;
    int i = idx & 15;
    float v = 0.0f;
    if (i < 10) {
      float d  = C_L * ((float)u * (1.0f / 128.0f)) - C_L * ((float)i * 0.1f);
      float d2 = d - C_L;
      float d3 = d + C_L;
      v = expf(-d * d * C_I4TAU) + expf(-d2 * d2 * C_I4TAU) + expf(-d3 * d3 * C_I4TAU);
    }
    A[idx] = v;
  }
}

// --------------------------------------------------------------------------
// Setup kernel 2: circular spectral operator m[10][10] from the 4 filter
// parameter vectors (F=64).  m absorbs deconv, the filter bank sum, both
// 1/nfft factors and the Re(IFFT·diag·FFT) composition.
// --------------------------------------------------------------------------
__global__ void build_M_kernel(const float* __restrict__ amp,
                               const float* __restrict__ beta,
                               const float* __restrict__ shift,
                               const float* __restrict__ hyper,
                               float* __restrict__ mOut)
{
  __shared__ float Dm[100];
  int tid = threadIdx.x;
  if (tid < 100) {
    int   sx = tid / 10, sy = tid % 10;
    float fx = (float)(sx - 5), fy = (float)(sy - 5);
    float kx = C_K0 * fx, ky = C_K0 * fy;
    float k2 = kx * kx + ky * ky;
    float kk = sqrtf(k2);
    float deconv = (C_PI / C_TAU) * expf(k2 * C_TAU);
    float tm = 0.0f;
    for (int f = 0; f < 64; ++f) {
      float a = amp[f], b = beta[f], s = shift[f], h = hyper[f];
      float ap = kk + s, am = kk - s;
      float p1 = a / (ap * ap + a * a);
      float p2 = a / (am * am + a * a);
      float fl = k2 / (k2 + 400.0f * h);
      tm += b * (p1 + p2) * fl;
    }
    Dm[tid] = deconv * tm;
  }
  __syncthreads();
  if (tid < 100) {
    int dx = tid / 10, dy = tid % 10;
    float acc = 0.0f;
    for (int s = 0; s < 100; ++s) {
      int   sx = s / 10, sy = s % 10;
      float fx = (float)(sx - 5), fy = (float)(sy - 5);
      float ang = C_W10 * (fx * (float)dx + fy * (float)dy);
      acc += Dm[s] * __cosf(ang);
    }
    mOut[tid] = acc * 1.0e-4f;
  }
}

// --------------------------------------------------------------------------
// Main fused kernel: one workgroup (8 wave32) per (b,c) image.
// LDS strides chosen for conflict-free WMMA operand fetches:
//   Xs stride 132 (132%64=4 -> 16 distinct banks down a column)
//   As/Ys stride 20 (20*r mod 64 distinct for r=0..15)
// --------------------------------------------------------------------------
__global__ void __launch_bounds__(256)
fourier_main(const float* __restrict__ x, const float* __restrict__ gA,
             const float* __restrict__ gm, float* __restrict__ out)
{
  __shared__ __align__(16) float Xs[128 * 132];   // image / final result
  __shared__ __align__(16) float As[128 * 20];    // A (128x16, padded)
  __shared__ __align__(16) float Ys[128 * 20];    // Y = X*A, then Z = A*T
  __shared__ __align__(16) float Sp[8 * 256];     // per-wave S partials
  __shared__ __align__(16) float Ss[256];         // S = A^T X A (16x16)
  __shared__ __align__(16) float Ts[256];         // T = m (*) S  (16x16)
  __shared__ __align__(16) float Ms[112];         // circular kernel m (10x10)

  const int tid  = threadIdx.x;
  const int wave = tid >> 5;
  const int lane = tid & 31;
  const int half = lane >> 4;     // lane half selects K (A/B) or M (C/D)
  const int l16  = lane & 15;
  const int img  = blockIdx.x;

  const float* __restrict__ xin  = x   + (size_t)img * 16384;
  float*       __restrict__ xout = out + (size_t)img * 16384;

  // ---- stage X into LDS: gfx1250 async HBM->LDS (no VGPR round-trip) ------
  {
    const v4f* __restrict__ x4 = (const v4f*)xin;
#if USE_ASYNC_LDS
#pragma unroll
    for (int it = 0; it < 16; ++it) {
      int idx = tid + it * 256;           // float4 index within image
      int row = idx >> 5;
      int c4  = (idx & 31) << 2;
      __builtin_amdgcn_global_load_async_to_lds_b128(
          (GV4i*)(x4 + idx), (LV4i*)&Xs[row * 132 + c4], 0, 0);
    }
#else
#pragma unroll
    for (int it = 0; it < 16; ++it) {
      int idx = tid + it * 256;
      int row = idx >> 5;
      int c4  = (idx & 31) << 2;
      *(v4f*)&Xs[row * 132 + c4] = x4[idx];
    }
#endif
  }

  // ---- stage A/m into LDS (overlaps with async X transfer) ----------------
  for (int idx = tid; idx < 2048; idx += 256)
    As[(idx >> 4) * 20 + (idx & 15)] = gA[idx];
  if (tid < 100) Ms[tid] = gm[tid];

#if USE_ASYNC_LDS
  WAIT_ASYNC0();                          // drain ASYNCcnt before barrier
#endif
  __syncthreads();

  // ---- Stage 2: Y = X * A   (128x128)*(128x16), one 16-row tile per wave --
  {
    const int m0 = wave << 4;
    v8f acc = {};
#pragma unroll 4
    for (int k0 = 0; k0 < 128; k0 += 4) {
      int kb = k0 + (half << 1);
      v2f a = *(const v2f*)&Xs[(m0 + l16) * 132 + kb];   // A-op: X rows
      v2f b;
      b[0] = As[kb * 20 + l16];                          // B-op: A rows
      b[1] = As[(kb + 1) * 20 + l16];
      acc = WMMA_F32(a, b, acc);
    }
#pragma unroll
    for (int r = 0; r < 8; ++r)
      Ys[(m0 + r + (half << 3)) * 20 + l16] = acc[r];
  }
  __syncthreads();

  // ---- Stage 3: S = A^T * Y  (16x128)*(128x16), K split across waves ------
  {
    const int kc = wave << 4;
    v8f acc = {};
#pragma unroll
    for (int k0 = 0; k0 < 16; k0 += 4) {
      int kg = kc + k0 + (half << 1);
      v2f a, b;
      a[0] = As[kg * 20 + l16];          // A-op: A^T  -> A column access
      a[1] = As[(kg + 1) * 20 + l16];
      b[0] = Ys[kg * 20 + l16];          // B-op: Y rows
      b[1] = Ys[(kg + 1) * 20 + l16];
      acc = WMMA_F32(a, b, acc);
    }
#pragma unroll
    for (int r = 0; r < 8; ++r)
      Sp[wave * 256 + (r + (half << 3)) * 16 + l16] = acc[r];
  }
  __syncthreads();

  // ---- Stage 4a: reduce the 8 partial S tiles ----------------------------
  {
    float s = 0.0f;
#pragma unroll
    for (int w = 0; w < 8; ++w) s += Sp[w * 256 + tid];
    Ss[tid] = s;
  }
  __syncthreads();

  // ---- Stage 4b: T = m (*) S   (circular conv on the 10x10 grid) ---------
  {
    int r = tid >> 4, c = tid & 15;
    float t = 0.0f;
    if (r < 10 && c < 10) {
      for (int i = 0; i < 10; ++i) {
        int dr = r - i; if (dr < 0) dr += 10;
        for (int j = 0; j < 10; ++j) {
          int dc = c - j; if (dc < 0) dc += 10;
          t += Ms[dr * 10 + dc] * Ss[i * 16 + j];
        }
      }
    }
    Ts[tid] = t;                         // zero outside 10x10
  }
  __syncthreads();

  // ---- Stage 5: Z = A * T   (128x16)*(16x16) -> Ys ------------------------
  {
    const int m0 = wave << 4;
    v8f acc = {};
#pragma unroll
    for (int k0 = 0; k0 < 16; k0 += 4) {
      int kg = k0 + (half << 1);
      v2f a = *(const v2f*)&As[(m0 + l16) * 20 + kg];    // A rows
      v2f b;
      b[0] = Ts[kg * 16 + l16];
      b[1] = Ts[(kg + 1) * 16 + l16];
      acc = WMMA_F32(a, b, acc);
    }
#pragma unroll
    for (int r = 0; r < 8; ++r)
      Ys[(m0 + r + (half << 3)) * 20 + l16] = acc[r];
  }
  __syncthreads();

  // ---- Stage 6: P = Z * A^T  (128x16)*(16x128) -> Xs (reuse) --------------
  {
    const int m0 = wave << 4;
#pragma unroll
    for (int nt = 0; nt < 8; ++nt) {
      const int n0 = nt << 4;
      v8f acc = {};
#pragma unroll
      for (int k0 = 0; k0 < 16; k0 += 4) {
        int kg = k0 + (half << 1);
        v2f a = *(const v2f*)&Ys[(m0 + l16) * 20 + kg];  // Z rows
        v2f b = *(const v2f*)&As[(n0 + l16) * 20 + kg];  // A^T -> A rows again
        acc = WMMA_F32(a, b, acc);
      }
#pragma unroll
      for (int r = 0; r < 8; ++r)
        Xs[(m0 + r + (half << 3)) * 132 + n0 + l16] = acc[r];
    }
  }
  __syncthreads();

  // ---- store result: async LDS->HBM (b128, coalesced) ---------------------
  {
    v4f* __restrict__ o4 = (v4f*)xout;
#if USE_ASYNC_LDS
#pragma unroll
    for (int it = 0; it < 16; ++it) {
      int idx = tid + it * 256;
      int row = idx >> 5;
      int c4  = (idx & 31) << 2;
      __builtin_amdgcn_global_store_async_from_lds_b128(
          (GV4i*)(o4 + idx), (LV4i*)&Xs[row * 132 + c4], 0, 0);
    }
    WAIT_ASYNC0();   // drain before wave exit (S_ENDPGM would also wait-idle)
#else
#pragma unroll
    for (int it = 0; it < 16; ++it) {
      int idx = tid + it * 256;
      int row = idx >> 5;
      int c4  = (idx & 31) << 2;
      o4[idx] = *(const v4f*)&Xs[row * 132 + c4];
    }
#endif
  }
}

// --------------------------------------------------------------------------
extern "C" void kernel_launch(void* const* d_in, const int* in_sizes, int n_in,
                              void* d_out, int out_size, void* d_ws, size_t ws_size,
                              hipStream_t stream)
{
  (void)in_sizes; (void)n_in; (void)out_size; (void)ws_size;
  const float* x     = (const float*)d_in[0];   // (16,128,128,128) f32
  const float* amp   = (const float*)d_in[1];   // (64,)
  const float* beta  = (const float*)d_in[2];   // (64,)
  const float* shift = (const float*)d_in[3];   // (64,)
  const float* hyper = (const float*)d_in[4];   // (64,)
  float* out = (float*)d_out;

  float* wsA = (float*)d_ws;     // 2048 floats: A matrix (128x16)
  float* wsM = wsA + 2048;       //  100 floats: circular kernel m

  build_A_kernel<<<1, 256, 0, stream>>>(wsA);
  build_M_kernel<<<1, 128, 0, stream>>>(amp, beta, shift, hyper, wsM);
  fourier_main<<<2048, 256, 0, stream>>>(x, wsA, wsM, out);
}